// STDP_19791209300726
// MI455X (gfx1250) — compile-verified
//
#include <hip/hip_runtime.h>

// STDP with AP=AN=1, TP=TN=2, TE=1:
//   TE=1  => eligibility has no memory: e = x_T (outer) post[T-1] - pre[T-1] (outer) y_T
//   TP=2  => x_T = sum_t pre[t] * 2^(t-T)   (geometric trace), same for y with post.
// => two weighted column-reductions + one rank-2 outer product (bandwidth bound,
//    ~84 MB of HBM traffic -> ~3.6 us floor at 23.3 TB/s).

typedef __attribute__((ext_vector_type(2))) float v2f;
typedef __attribute__((ext_vector_type(8))) float v8f;

constexpr int T      = 512;
constexpr int L      = 4096;       // L1 == L2 == 4096
constexpr int NCHUNK = 4;
constexpr int CT     = T / NCHUNK; // 128
constexpr int NT     = 4;          // 16-col tiles per wave in the WMMA kernel

// ---------------------------------------------------------------------------
// Phase 1: per-chunk local recurrence v <- v/2 + src[t]/2 (starts from 0).
// partial[a][c][i] = sum_{t in chunk c} src[t][i] * 2^(t - chunk_end)
// grid: (L/256, NCHUNK, 2)   a=0 -> pre, a=1 -> post
// ---------------------------------------------------------------------------
__global__ void stdp_trace_partial(const float* __restrict__ pre,
                                   const float* __restrict__ post,
                                   float* __restrict__ partial) {
  const int i = blockIdx.x * blockDim.x + threadIdx.x;  // column
  const int c = blockIdx.y;                             // time chunk
  const int a = blockIdx.z;                             // which array
  const float* __restrict__ src = (a == 0) ? pre : post;
  const int t0 = c * CT;
  float v = 0.0f;
#pragma unroll 8
  for (int t = 0; t < CT; ++t) {
    v = fmaf(0.5f, v, 0.5f * src[(size_t)(t0 + t) * L + i]);
  }
  partial[((size_t)a * NCHUNK + c) * L + i] = v;
}

// ---------------------------------------------------------------------------
// Phase 2: combine chunks with exact power-of-two scaling (fixed order =>
// deterministic). xy[0..L) = x trace, xy[L..2L) = y trace.
// ---------------------------------------------------------------------------
__global__ void stdp_trace_combine(const float* __restrict__ partial,
                                   float* __restrict__ xy) {
  const int g   = blockIdx.x * blockDim.x + threadIdx.x;  // 0 .. 2L-1
  const int a   = (g >= L) ? 1 : 0;
  const int col = g - a * L;
  float s = 0.0f;
#pragma unroll
  for (int c = 0; c < NCHUNK; ++c) {
    const float pv = partial[((size_t)a * NCHUNK + c) * L + col];
    s += ldexpf(pv, (c + 1) * CT - T);   // scale by 2^(chunk_end - T)
  }
  xy[g] = s;
}

// ---------------------------------------------------------------------------
// Phase 3: e[i][j] = x[i]*q[j] - p[i]*y[j] as 16x16 tiles via
// V_WMMA_F32_16X16X4_F32 with A = [x, -p, 0, 0] (16x4), B = [q; y; 0; 0] (4x16).
// A layout (ISA 7.12.2): lanes 0-15 hold K=0 (a.x) and K=1 (a.y); lanes 16-31
// hold K=2/3 -> zeroed. B rows K=0/1 likewise live in lanes 0-15. C/D: lane<16
// -> N=lane, M=vgpr; lane>=16 -> N=lane-16, M=vgpr+8.
// One wave emits NT=4 tiles (A operand reused). EXEC stays all-ones: loads are
// unconditional, zeroing is done with selects.
// ---------------------------------------------------------------------------
__global__ void stdp_outer_wmma(const float* __restrict__ xy,
                                const float* __restrict__ pre_last,
                                const float* __restrict__ post_last,
                                float* __restrict__ out) {
  const int lane          = threadIdx.x & 31;
  const int wave          = threadIdx.x >> 5;
  const int wavesPerBlock = blockDim.x >> 5;
  const int group         = blockIdx.x * wavesPerBlock + wave;
  const int groupsPerRow  = (L / 16) / NT;          // 64
  const int tm            = group / groupsPerRow;   // row tile
  const int tn0           = (group % groupsPerRow) * NT;

  const int  l  = lane & 15;
  const bool lo = lane < 16;

  const int   row = tm * 16 + l;
  const float xv  = xy[row];
  const float pv  = pre_last[row];
  v2f A;
  A.x = lo ?  xv : 0.0f;   // K=0 column of A
  A.y = lo ? -pv : 0.0f;   // K=1 column of A (negation folded in)

  const float* __restrict__ ytr = xy + L;

#pragma unroll
  for (int nt = 0; nt < NT; ++nt) {
    const int   col = (tn0 + nt) * 16 + l;
    const float qv  = post_last[col];
    const float yv  = ytr[col];
    v2f B;
    B.x = lo ? qv : 0.0f;  // K=0 row of B
    B.y = lo ? yv : 0.0f;  // K=1 row of B

    v8f C = {};
    C = __builtin_amdgcn_wmma_f32_16x16x4_f32(
        /*neg_a=*/false, A, /*neg_b=*/false, B,
        /*c_mod=*/(short)0, C, /*reuse_a=*/false, /*reuse_b=*/false);

    const int rbase = tm * 16 + (lo ? 0 : 8);
    float* __restrict__ o = out + (size_t)rbase * L + col;
#pragma unroll
    for (int v = 0; v < 8; ++v) {
      o[(size_t)v * L] = C[v];
    }
  }
}

// ---------------------------------------------------------------------------
extern "C" void kernel_launch(void* const* d_in, const int* in_sizes, int n_in,
                              void* d_out, int out_size, void* d_ws, size_t ws_size,
                              hipStream_t stream) {
  (void)in_sizes; (void)n_in; (void)out_size; (void)ws_size;

  const float* pre  = (const float*)d_in[0];   // [T, L]
  const float* post = (const float*)d_in[1];   // [T, L]
  float*       out  = (float*)d_out;           // [L, L]

  float* partial = (float*)d_ws;               // 2 * NCHUNK * L floats (128 KB)
  float* xy      = partial + 2 * NCHUNK * L;   // 2 * L floats (32 KB)

  // Phase 1: chunked trace recurrences (32768 threads, coalesced row reads).
  dim3 g1(L / 256, NCHUNK, 2);
  stdp_trace_partial<<<g1, dim3(256), 0, stream>>>(pre, post, partial);

  // Phase 2: deterministic chunk combine.
  stdp_trace_combine<<<(2 * L) / 256, 256, 0, stream>>>(partial, xy);

  // Phase 3: rank-2 outer product via f32 WMMA. (L/16)^2 tiles, NT per wave.
  const int groups        = (L / 16) * ((L / 16) / NT);  // 16384
  const int wavesPerBlock = 8;                           // 256 threads/block
  stdp_outer_wmma<<<groups / wavesPerBlock, 256, 0, stream>>>(
      xy,
      pre  + (size_t)(T - 1) * L,
      post + (size_t)(T - 1) * L,
      out);
}